// CollapseLoss_41317585387602
// MI455X (gfx1250) — compile-verified
//
#include <hip/hip_runtime.h>
#include <hip/hip_bf16.h>

typedef __attribute__((ext_vector_type(2))) float v2f;
typedef __attribute__((ext_vector_type(8))) float v8f;

#define N_ROWS 16384
#define DIMS   128
#define EPS_N  1e-12f

// ---------------------------------------------------------------------------
// Pass 1: inv[i] = 1 / max(||fc[i]||_2, eps), one wave32 per row.
// Each lane loads float4 (lane*4 .. lane*4+3), 32 lanes * 4 = 128 floats.
// ---------------------------------------------------------------------------
__global__ void collapse_inv_norm_kernel(const float* __restrict__ fc,
                                         float* __restrict__ inv) {
  const int wave = (blockIdx.x * blockDim.x + threadIdx.x) >> 5;
  const int lane = threadIdx.x & 31;
  const float4* p = (const float4*)(fc + (size_t)wave * DIMS) + lane;
  float4 v = *p;
  float ss = v.x * v.x + v.y * v.y + v.z * v.z + v.w * v.w;
#pragma unroll
  for (int off = 16; off > 0; off >>= 1) {
    ss += __shfl_xor(ss, off, 32);
  }
  if (lane == 0) {
    inv[wave] = 1.0f / fmaxf(sqrtf(ss), EPS_N);
  }
}

// ---------------------------------------------------------------------------
// Pass 2: one wave32 per 16-output tile.
// D = A x B with A rows = fc[base..base+15], B cols = fc[base+1..base+16].
// diag(D)[m] = dot(fc[base+m], fc[base+m+1]).  K=128 via 32x WMMA F32 16x16x4.
//
// f32 WMMA operand layout (wave32):
//   A 16x4: lane L -> M = L%16, K = 2*(L/16) + {0,1} in vgpr {0,1}
//   B 4x16: lane L -> N = L%16, K = 2*(L/16) + {0,1} in vgpr {0,1}
// => both operands are a per-lane float2 load at row*128 + k + 2*(L>>4),
//    with B's row shifted by +1.
// C/D 16x16 f32: vgpr r: lanes 0-15 -> (M=r, N=lane); lanes 16-31 -> (M=8+r,
// N=lane-16).  Diagonal (m,m): lane m (m<8, vgpr m) and lane m+16 (m>=8,
// vgpr m-8), i.e. lanes 0-7 and 24-31.
// ---------------------------------------------------------------------------
__global__ void collapse_adj_dot_kernel(const float* __restrict__ fc,
                                        const float* __restrict__ inv,
                                        float* __restrict__ out) {
  const int lane = threadIdx.x & 31;
  const int wavesPerBlk = blockDim.x >> 5;
  const int tile = blockIdx.x * wavesPerBlk + (threadIdx.x >> 5);
  const int base = tile * 16;

  const int m    = lane & 15;   // M for A / N for B
  const int half = lane >> 4;   // selects K pair {0,1} vs {2,3}

  const int rowA = base + m;
  int rowB = base + 1 + m;
  if (rowB > N_ROWS - 1) rowB = N_ROWS - 1;  // last tile: d[16383] unused

  const float* pa = fc + (size_t)rowA * DIMS + 2 * half;
  const float* pb = fc + (size_t)rowB * DIMS + 2 * half;

  v8f acc = {0.f, 0.f, 0.f, 0.f, 0.f, 0.f, 0.f, 0.f};
#pragma unroll
  for (int k = 0; k < DIMS; k += 4) {
    v2f a = *(const v2f*)(pa + k);   // 8B aligned: row base 512B + 8*half
    v2f b = *(const v2f*)(pb + k);
    // (neg_a, A, neg_b, B, c_mod, C, reuse_a, reuse_b)
    acc = __builtin_amdgcn_wmma_f32_16x16x4_f32(
        false, a, false, b, (short)0, acc, false, false);
  }

  // Extract the diagonal held by this lane (lanes 0-7 and 24-31 only).
  float diag = 0.f;
#pragma unroll
  for (int r = 0; r < 8; ++r) {
    float cr = acc[r];
    if (lane == r)      diag = cr;  // (M=r,     N=r)
    if (lane == r + 24) diag = cr;  // (M=8+r,   N=8+r)
  }

  const bool have = (lane < 8) || (lane >= 24);
  const int  md   = (lane < 8) ? lane : (lane - 16);  // diagonal index m
  const int  oidx = base + md;
  if (have && oidx < N_ROWS - 1) {
    float d = diag * inv[oidx] * inv[oidx + 1];
    const float c = 1.0f / (float)(N_ROWS - 1);
    float t = d + c;
    out[oidx] = t * t;
  }
}

// ---------------------------------------------------------------------------
extern "C" void kernel_launch(void* const* d_in, const int* in_sizes, int n_in,
                              void* d_out, int out_size, void* d_ws, size_t ws_size,
                              hipStream_t stream) {
  (void)in_sizes; (void)n_in; (void)out_size; (void)ws_size;
  const float* fc  = (const float*)d_in[0];
  float*       out = (float*)d_out;
  float*       inv = (float*)d_ws;  // 16384 floats = 64 KB scratch

  // Pass 1: 16384 rows * 32 lanes / 256 threads = 2048 blocks
  collapse_inv_norm_kernel<<<(N_ROWS * 32) / 256, 256, 0, stream>>>(fc, inv);

  // Pass 2: 1024 tiles, 8 waves (tiles) per 256-thread block = 128 blocks
  const int tiles = N_ROWS / 16;  // 1024
  collapse_adj_dot_kernel<<<tiles / 8, 256, 0, stream>>>(fc, inv, out);
}